// OrthogonalityRegularization_5162550689875
// MI455X (gfx1250) — compile-verified
//
#include <hip/hip_runtime.h>

// ---------------------------------------------------------------------------
// OrthogonalityRegularization, fully collapsed:
//   out = ( || colsum ||^2 - sum(x^4) ) / D,   colsum[d] = sum_b x[b,d]^2
// Single streaming pass over 48 MiB  (~2.2 us at 23.3 TB/s HBM).
// The cross-row reduction runs on the CDNA5 matrix pipe:
//   V_WMMA_F32_16X16X4_F32 with A = ones(16x4) sums 4 rows x 16 cols per op,
//   accumulating in fp32 in the C operand.
// ---------------------------------------------------------------------------

typedef __attribute__((ext_vector_type(2))) float v2f;
typedef __attribute__((ext_vector_type(8))) float v8f;

#define B_ROWS 1024
#define D_COLS 12288
#define ROW_CHUNKS 8
#define ROWS_PER_CHUNK (B_ROWS / ROW_CHUNKS)      // 128
#define COL_TILES (D_COLS / 16)                   // 768
#define WAVES_PER_BLOCK 8
#define THREADS_PER_BLOCK (WAVES_PER_BLOCK * 32)  // 256
#define TOTAL_WAVES (COL_TILES * ROW_CHUNKS)      // 6144
#define N_BLOCKS (TOTAL_WAVES / WAVES_PER_BLOCK)  // 768
#define WS_FLOATS (D_COLS + 1)                    // colsum[12288] + term2

__global__ void zero_ws_kernel(float* __restrict__ ws, int n) {
    int i = blockIdx.x * blockDim.x + threadIdx.x;
    if (i < n) ws[i] = 0.0f;
}

__global__ __launch_bounds__(THREADS_PER_BLOCK)
void colsum_wmma_kernel(const float* __restrict__ X, float* __restrict__ ws) {
    const int wave    = (blockIdx.x * THREADS_PER_BLOCK + threadIdx.x) >> 5;
    const int lane    = threadIdx.x & 31;
    const int colTile = wave % COL_TILES;
    const int chunk   = wave / COL_TILES;

    // B-matrix (4x16 fp32, 2 VGPRs) lane mapping:
    //   lanes 0-15 : n = lane,    v[0] = K0 row, v[1] = K1 row
    //   lanes 16-31: n = lane-16, v[0] = K2 row, v[1] = K3 row
    const int n    = lane & 15;
    const int half = lane >> 4;
    const int col  = colTile * 16 + n;
    const int row0 = chunk * ROWS_PER_CHUNK + 2 * half;

    const float* p = X + (size_t)row0 * D_COLS + col;

    v8f c = {};                 // fp32 accumulator (C/D 16x16)
    v2f a;                      // all-ones A (16x4): every lane {1,1}
    a.x = 1.0f; a.y = 1.0f;
    float t2 = 0.0f;            // per-lane sum of x^4

    #pragma unroll 4
    for (int s = 0; s < ROWS_PER_CHUNK / 4; ++s) {
        float x0 = p[0];
        float x1 = p[D_COLS];
        __builtin_prefetch(p + 8 * D_COLS, 0, 1);   // global_prefetch_b8 ahead
        p += 4 * D_COLS;

        float s0 = x0 * x0;
        float s1 = x1 * x1;
        t2 += s0 * s0 + s1 * s1;

        v2f b; b.x = s0; b.y = s1;
        // C += ones(16x4) x B(4x16)  -> every row of C = per-column sum
        c = __builtin_amdgcn_wmma_f32_16x16x4_f32(
                /*neg_a=*/false, a, /*neg_b=*/false, b,
                /*c_mod=*/(short)0, c, /*reuse_a=*/false, /*reuse_b=*/false);
    }

    // C VGPR0: lanes 0-15 hold M=0,N=lane ; lanes 16-31 duplicate (M=8).
    // Only the low half-wave commits the partial column sums.
    if (lane < 16) {
        atomicAdd(&ws[col], c[0]);
    }

    // Wave-reduce the x^4 partial (wave32) and commit once per wave.
    for (int off = 16; off > 0; off >>= 1)
        t2 += __shfl_xor(t2, off, 32);
    if (lane == 0) {
        atomicAdd(&ws[D_COLS], t2);
    }
}

__global__ __launch_bounds__(1024)
void finalize_kernel(const float* __restrict__ ws, float* __restrict__ out) {
    __shared__ float red[32];
    const int tid  = threadIdx.x;
    const int lane = tid & 31;
    const int w    = tid >> 5;

    float acc = 0.0f;
    for (int d = tid; d < D_COLS; d += 1024) {
        float cs = ws[d];
        acc += cs * cs;
    }
    for (int off = 16; off > 0; off >>= 1)
        acc += __shfl_xor(acc, off, 32);
    if (lane == 0) red[w] = acc;
    __syncthreads();

    if (w == 0) {
        float v = (lane < 32) ? red[lane] : 0.0f;
        for (int off = 16; off > 0; off >>= 1)
            v += __shfl_xor(v, off, 32);
        if (lane == 0) {
            out[0] = (v - ws[D_COLS]) / (float)D_COLS;
        }
    }
}

extern "C" void kernel_launch(void* const* d_in, const int* in_sizes, int n_in,
                              void* d_out, int out_size, void* d_ws, size_t ws_size,
                              hipStream_t stream) {
    const float* X  = (const float*)d_in[0];
    float* out      = (float*)d_out;
    float* ws       = (float*)d_ws;

    (void)in_sizes; (void)n_in; (void)out_size; (void)ws_size;

    zero_ws_kernel<<<(WS_FLOATS + 255) / 256, 256, 0, stream>>>(ws, WS_FLOATS);
    colsum_wmma_kernel<<<N_BLOCKS, THREADS_PER_BLOCK, 0, stream>>>(X, ws);
    finalize_kernel<<<1, 1024, 0, stream>>>(ws, out);
}